// Net_3676492005914
// MI455X (gfx1250) — compile-verified
//
#include <hip/hip_runtime.h>
#include <hip/hip_bf16.h>

#define NN   65536      // total nodes
#define HH   128        // hidden / F_IN
#define EE   1048576    // edges
#define BB   256        // graphs
#define NPER 256        // nodes per graph
#define ASTR 132        // padded LDS row stride (floats): conflict-free, 16B-aligned

typedef float v2f __attribute__((ext_vector_type(2)));
typedef float v8f __attribute__((ext_vector_type(8)));

// ---- async global->LDS copy (gfx1250 GLOBAL_LOAD_ASYNC_TO_LDS_B128), guarded ----
#if defined(__has_builtin)
#if __has_builtin(__builtin_amdgcn_global_load_async_to_lds_b128) && \
    __has_builtin(__builtin_amdgcn_s_wait_asynccnt)
#define HAVE_ASYNC 1
#endif
#endif
#ifndef HAVE_ASYNC
#define HAVE_ASYNC 0
#endif

typedef int vi4 __attribute__((vector_size(16)));              // V4i
typedef __attribute__((address_space(1))) vi4 gvi4;            // global V4i
typedef __attribute__((address_space(3))) vi4 lvi4;            // LDS V4i

__device__ __forceinline__ void async_cp16(const float* g, float* l) {
#if HAVE_ASYNC
    __builtin_amdgcn_global_load_async_to_lds_b128((gvi4*)g, (lvi4*)l, 0, 0);
#else
    *(float4*)l = *(const float4*)g;
#endif
}
__device__ __forceinline__ void async_wait0() {
#if HAVE_ASYNC
    __builtin_amdgcn_s_wait_asynccnt(0);
#endif
}

// ---------------- utility fills ----------------
__global__ void fill_f32(float* p, float v, int n) {
    int i = blockIdx.x * blockDim.x + threadIdx.x;
    if (i < n) p[i] = v;
}
__global__ void fill_i32(int* p, int v, int n) {
    int i = blockIdx.x * blockDim.x + threadIdx.x;
    if (i < n) p[i] = v;
}

// ---------------- deterministic CSR build (dst-major) ----------------
__global__ void csr_count(const int* __restrict__ dst, int* __restrict__ cnt) {
    int e = blockIdx.x * blockDim.x + threadIdx.x;
    if (e < EE) atomicAdd(&cnt[dst[e]], 1);
}
__global__ __launch_bounds__(256) void csr_scan(const int* __restrict__ cnt, int* __restrict__ offs) {
    __shared__ int s[NPER];
    const int g = blockIdx.x, t = threadIdx.x;
    const int c = cnt[g * NPER + t];
    s[t] = c;
    __syncthreads();
    for (int d = 1; d < NPER; d <<= 1) {
        int v = (t >= d) ? s[t - d] : 0;
        __syncthreads();
        s[t] += v;
        __syncthreads();
    }
    offs[g * NPER + t] = g * (NPER * 16) + s[t] - c;   // 4096 edges per graph
}
__global__ void csr_fill(const int* __restrict__ dst, int* __restrict__ cur,
                         const int* __restrict__ offs, int* __restrict__ eid) {
    int e = blockIdx.x * blockDim.x + threadIdx.x;
    if (e >= EE) return;
    int d = dst[e];
    int p = atomicAdd(&cur[d], 1);
    eid[offs[d] + p] = e;
}
// sort each bucket by edge id -> fully deterministic accumulation order
__global__ void csr_sort(int* __restrict__ eid, const int* __restrict__ offs,
                         const int* __restrict__ cnt) {
    int n = blockIdx.x * blockDim.x + threadIdx.x;
    if (n >= NN) return;
    int o = offs[n], c = cnt[n];
    for (int i = 1; i < c; ++i) {
        int v = eid[o + i], j = i - 1;
        while (j >= 0 && eid[o + j] > v) { eid[o + j + 1] = eid[o + j]; --j; }
        eid[o + j + 1] = v;
    }
}

// ---------------- per-layer GCN pieces ----------------
__global__ void deg_dinv(const float* __restrict__ maskf, const int* __restrict__ srcA,
                         const int* __restrict__ eid, const int* __restrict__ offs,
                         const int* __restrict__ cnt, float* __restrict__ dinv) {
    int n = blockIdx.x * blockDim.x + threadIdx.x;
    if (n >= NN) return;
    float m = maskf[n];
    if (m > 0.f) {
        float s = 1.f;
        int o = offs[n], c = cnt[n];
        for (int j = 0; j < c; ++j) s += maskf[srcA[eid[o + j]]];
        dinv[n] = rsqrtf(s);
    } else {
        dinv[n] = 0.f;
    }
}

// Out[N,128] = X[N,128] @ W[128,128] via V_WMMA_F32_16X16X4_F32 (fp32-exact).
// Block = 8 waves, 64-row super-tile: A staged async into padded LDS (bank-
// conflict-free ds_load_b64 fragments), per-wave B tile preloaded to VGPRs and
// reused across 4 M-tiles -> inner loop is LDS + WMMA only.
__global__ __launch_bounds__(256) void gemm128_wmma(const float* __restrict__ X,
                                                    const float* __restrict__ W,
                                                    float* __restrict__ Out) {
    __shared__ float As[64 * ASTR];        // ~33 KB
    const int tid = threadIdx.x;
    const int m0  = blockIdx.x * 64;
    // stage 64 rows of A: 2048 16B chunks, 8 per thread, coalesced, async
    const float* Xb = X + (size_t)m0 * HH;
    #pragma unroll
    for (int i = 0; i < 8; ++i) {
        int c   = tid + i * 256;           // chunk id: row = c/32, col16 = c%32
        int row = c >> 5, col = c & 31;
        async_cp16(Xb + c * 4, As + row * ASTR + col * 4);
    }

    const int lane = tid & 31;
    const int n0   = (tid >> 5) * 16;      // each of 8 waves owns a 16-wide N tile
    const int half = lane >> 4;
    const int l    = lane & 15;

    // preload this wave's B fragments (W[:, n0:n0+16]) into 64 VGPRs
    v2f wf[32];
    #pragma unroll
    for (int k4 = 0; k4 < 32; ++k4) {
        const int ka = 4 * k4 + 2 * half;
        wf[k4].x = W[(size_t)ka * HH + n0 + l];
        wf[k4].y = W[(size_t)(ka + 1) * HH + n0 + l];
    }

    async_wait0();
    __syncthreads();

    #pragma unroll
    for (int mt = 0; mt < 4; ++mt) {
        const float* Ar = As + (mt * 16 + l) * ASTR + 2 * half;
        v8f acc = {};
        #pragma unroll
        for (int k4 = 0; k4 < 32; ++k4) {
            v2f a;
            a.x = Ar[4 * k4];
            a.y = Ar[4 * k4 + 1];
            acc = __builtin_amdgcn_wmma_f32_16x16x4_f32(false, a, false, wf[k4],
                                                        (short)0, acc, false, false);
        }
        const int row = m0 + mt * 16 + half * 8;
        #pragma unroll
        for (int j = 0; j < 8; ++j)
            Out[(size_t)(row + j) * HH + n0 + l] = acc[j];
    }
}

// Out[n] = relu?( b + dinv[n]*(dinv[n]*HW[n] + sum_in dinv[s]*HW[s]) )
__global__ __launch_bounds__(128) void agg_kernel(const float* __restrict__ HW,
        const float* __restrict__ bias, const float* __restrict__ dinv,
        const int* __restrict__ srcA, const int* __restrict__ eid,
        const int* __restrict__ offs, const int* __restrict__ cnt,
        float* __restrict__ Out, int doRelu) {
    const int n = blockIdx.x;
    const int f = threadIdx.x;
    const float dn = dinv[n];
    float acc = dn * HW[(size_t)n * HH + f];
    const int o = offs[n], c = cnt[n];
    for (int j = 0; j < c; ++j) {
        const int s = srcA[eid[o + j]];
        acc += dinv[s] * HW[(size_t)s * HH + f];
    }
    float r = bias[f] + dn * acc;
    if (doRelu) r = fmaxf(r, 0.f);
    Out[(size_t)n * HH + f] = r;
}

// scw[n] = dot(H[n,:], Ws[:,0])  -- one wave32 per node
__global__ __launch_bounds__(256) void score_mv(const float* __restrict__ Hf,
        const float* __restrict__ Ws, float* __restrict__ scw) {
    const int t = threadIdx.x;
    const int lane = t & 31;
    const int n = blockIdx.x * 8 + (t >> 5);
    const float* hp = Hf + (size_t)n * HH;
    float a = hp[lane] * Ws[lane] + hp[lane + 32] * Ws[lane + 32]
            + hp[lane + 64] * Ws[lane + 64] + hp[lane + 96] * Ws[lane + 96];
    for (int off = 16; off > 0; off >>= 1) a += __shfl_down(a, off);
    if (lane == 0) scw[n] = a;
}

__global__ void score_agg(const float* __restrict__ scw, const float* __restrict__ bs,
        const float* __restrict__ dinv, const int* __restrict__ srcA,
        const int* __restrict__ eid, const int* __restrict__ offs,
        const int* __restrict__ cnt, float* __restrict__ score) {
    int n = blockIdx.x * blockDim.x + threadIdx.x;
    if (n >= NN) return;
    const float dn = dinv[n];
    float acc = dn * scw[n];
    int o = offs[n], c = cnt[n];
    for (int j = 0; j < c; ++j) {
        int s = srcA[eid[o + j]];
        acc += dinv[s] * scw[s];
    }
    score[n] = bs[0] + dn * acc;
}

// jax.lax.top_k semantics: rank = #(strictly greater) + #(equal with smaller index)
__global__ __launch_bounds__(256) void topk_kernel(const float* __restrict__ score,
        float* __restrict__ maskf, int k) {
    __shared__ float ls[NPER];
    const int g = blockIdx.x, t = threadIdx.x;
    const int n = g * NPER + t;
    const float s = (maskf[n] > 0.f) ? score[n] : -__builtin_inff();
    ls[t] = s;
    __syncthreads();
    int r = 0;
    for (int j = 0; j < NPER; ++j) {
        const float sj = ls[j];
        r += (sj > s) || (sj == s && j < t);
    }
    maskf[n] = (r < k) ? 1.f : 0.f;
}

__global__ void apply_pool(float* __restrict__ Hf, const float* __restrict__ score,
                           const float* __restrict__ maskf) {
    int i = blockIdx.x * blockDim.x + threadIdx.x;
    if (i >= NN * HH) return;
    int n = i >> 7;
    Hf[i] *= tanhf(score[n]) * maskf[n];
}

// r[g] += [max-pool || mean-pool] over surviving nodes
__global__ __launch_bounds__(128) void readout_kernel(const float* __restrict__ Hf,
        const float* __restrict__ maskf, float* __restrict__ rbuf, float kf) {
    const int g = blockIdx.x, f = threadIdx.x;
    float mx = -__builtin_inff(), sm = 0.f;
    for (int i = 0; i < NPER; ++i) {
        if (maskf[g * NPER + i] > 0.f) {
            float v = Hf[(size_t)(g * NPER + i) * HH + f];
            mx = fmaxf(mx, v);
            sm += v;
        }
    }
    rbuf[g * 2 * HH + f]      += mx;
    rbuf[g * 2 * HH + HH + f] += sm / kf;
}

// ---------------- MLP head ----------------
__global__ void mlp1_kernel(const float* __restrict__ r, const float* __restrict__ W,
                            const float* __restrict__ b, float* __restrict__ z) {
    int i = blockIdx.x * blockDim.x + threadIdx.x;      // i = bg*128 + j
    if (i >= BB * HH) return;
    int bg = i >> 7, j = i & 127;
    float a = b[j];
    for (int k = 0; k < 2 * HH; ++k) a += r[bg * 2 * HH + k] * W[k * HH + j];
    z[i] = fmaxf(a, 0.f);
}
__global__ void mlp2_kernel(const float* __restrict__ z1, const float* __restrict__ W,
                            const float* __restrict__ b, float* __restrict__ z2) {
    int i = blockIdx.x * blockDim.x + threadIdx.x;      // i = bg*64 + j
    if (i >= BB * 64) return;
    int bg = i >> 6, j = i & 63;
    float a = b[j];
    for (int k = 0; k < HH; ++k) a += z1[bg * HH + k] * W[k * 64 + j];
    z2[i] = fmaxf(a, 0.f);
}
__global__ void head_kernel(const float* __restrict__ z2, const float* __restrict__ W,
                            const float* __restrict__ b, float* __restrict__ out) {
    int bg = blockIdx.x * blockDim.x + threadIdx.x;
    if (bg >= BB) return;
    float t[10];
    for (int c = 0; c < 10; ++c) {
        float a = b[c];
        for (int k = 0; k < 64; ++k) a += z2[bg * 64 + k] * W[k * 10 + c];
        t[c] = fmaxf(a, 0.f);
    }
    float m = t[0];
    for (int c = 1; c < 10; ++c) m = fmaxf(m, t[c]);
    float se = 0.f;
    for (int c = 0; c < 10; ++c) se += expf(t[c] - m);
    float lse = logf(se);
    for (int c = 0; c < 10; ++c) out[bg * 10 + c] = t[c] - m - lse;
}

// ---------------- host side ----------------
extern "C" void kernel_launch(void* const* d_in, const int* in_sizes, int n_in,
                              void* d_out, int out_size, void* d_ws, size_t ws_size,
                              hipStream_t stream) {
    const float* x   = (const float*)d_in[0];
    const int*   src = (const int*)  d_in[1];
    const int*   dst = (const int*)  d_in[2];
    const float* Wc[3]  = { (const float*)d_in[3], (const float*)d_in[5], (const float*)d_in[7] };
    const float* bc[3]  = { (const float*)d_in[4], (const float*)d_in[6], (const float*)d_in[8] };
    const float* Wsc[3] = { (const float*)d_in[9],  (const float*)d_in[11], (const float*)d_in[13] };
    const float* bsc[3] = { (const float*)d_in[10], (const float*)d_in[12], (const float*)d_in[14] };
    const float* Wl1 = (const float*)d_in[15]; const float* bl1 = (const float*)d_in[16];
    const float* Wl2 = (const float*)d_in[17]; const float* bl2 = (const float*)d_in[18];
    const float* Wl3 = (const float*)d_in[19]; const float* bl3 = (const float*)d_in[20];
    float* out = (float*)d_out;

    // workspace layout
    float* bufA  = (float*)d_ws;                 // N*H  node features
    float* bufB  = bufA  + (size_t)NN * HH;      // N*H  h @ W
    float* dinv  = bufB  + (size_t)NN * HH;      // N
    float* scw   = dinv  + NN;                   // N
    float* score = scw   + NN;                   // N
    float* maskf = score + NN;                   // N
    float* rbuf  = maskf + NN;                   // B*2H
    float* z1    = rbuf  + BB * 2 * HH;          // B*H
    float* z2    = z1    + BB * HH;              // B*64
    int*   cnt   = (int*)(z2 + BB * 64);         // N
    int*   offs  = cnt  + NN;                    // N
    int*   cur   = offs + NN;                    // N
    int*   eid   = cur  + NN;                    // E

    const int T = 256;
    // deterministic CSR (structure is layer-invariant)
    fill_i32<<<NN / T, T, 0, stream>>>(cnt, 0, NN);
    fill_i32<<<NN / T, T, 0, stream>>>(cur, 0, NN);
    csr_count<<<EE / T, T, 0, stream>>>(dst, cnt);
    csr_scan<<<BB, NPER, 0, stream>>>(cnt, offs);
    csr_fill<<<EE / T, T, 0, stream>>>(dst, cur, offs, eid);
    csr_sort<<<NN / T, T, 0, stream>>>(eid, offs, cnt);

    fill_f32<<<NN / T, T, 0, stream>>>(maskf, 1.f, NN);
    fill_f32<<<(BB * 2 * HH) / T, T, 0, stream>>>(rbuf, 0.f, BB * 2 * HH);

    const int K[3] = { 128, 64, 32 };
    const float* hin = x;
    for (int L = 0; L < 3; ++L) {
        deg_dinv<<<NN / T, T, 0, stream>>>(maskf, src, eid, offs, cnt, dinv);
        gemm128_wmma<<<NN / 64, 256, 0, stream>>>(hin, Wc[L], bufB);
        agg_kernel<<<NN, HH, 0, stream>>>(bufB, bc[L], dinv, src, eid, offs, cnt, bufA, 1);
        score_mv<<<NN / 8, 256, 0, stream>>>(bufA, Wsc[L], scw);
        score_agg<<<NN / T, T, 0, stream>>>(scw, bsc[L], dinv, src, eid, offs, cnt, score);
        topk_kernel<<<BB, NPER, 0, stream>>>(score, maskf, K[L]);
        apply_pool<<<(NN * HH) / T, T, 0, stream>>>(bufA, score, maskf);
        readout_kernel<<<BB, HH, 0, stream>>>(bufA, maskf, rbuf, (float)K[L]);
        hin = bufA;
    }
    mlp1_kernel<<<(BB * HH) / T, T, 0, stream>>>(rbuf, Wl1, bl1, z1);
    mlp2_kernel<<<(BB * 64) / T, T, 0, stream>>>(z1, Wl2, bl2, z2);
    head_kernel<<<1, BB, 0, stream>>>(z2, Wl3, bl3, out);
}